// DistanceTransform_75239237091934
// MI455X (gfx1250) — compile-verified
//
#include <hip/hip_runtime.h>
#include <math.h>
#include <stdint.h>

// Exact Euclidean distance transform (Meijster), 1536x1536.
// Phase 1 (vertical scans) split into 3 passes for parallelism via min-plus
// linearity of the recurrence. Phase 2 uses an exact adaptive outward search
// per pixel over a g^2 row staged in LDS with gfx1250 async-to-LDS copies.

#define RR 1536
#define CC 1536
#define SEG 32
#define NSEG 48          // SEG * NSEG == RR
#define TPB 256
#define BIGF 3072.0f     // R + C, matches reference sentinel
#define INF_F 3.0e38f

// ---- Phase 1, pass A: per-(column, segment) local scans; emit carries only.
__global__ __launch_bounds__(TPB)
void edt_passA(const float* __restrict__ mask,
               float* __restrict__ carryD, float* __restrict__ carryU) {
    int col = blockIdx.x * TPB + threadIdx.x;
    int seg = blockIdx.y;
    if (col >= CC) return;
    int s0 = seg * SEG;

    float d = BIGF;                       // incoming assumed BIG (fixed in pass C)
    for (int r = s0; r < s0 + SEG; ++r) {
        float m = mask[r * CC + col];
        d = (m == 0.0f) ? 0.0f : fminf(d + 1.0f, BIGF);
    }
    carryD[seg * CC + col] = d;           // down-value at segment's last row

    float u = BIGF;
    for (int r = s0 + SEG - 1; r >= s0; --r) {
        float m = mask[r * CC + col];
        u = (m == 0.0f) ? 0.0f : fminf(u + 1.0f, BIGF);
    }
    carryU[seg * CC + col] = u;           // up-value at segment's first row
}

// ---- Phase 1, pass B: tiny per-column scan over NSEG carries -> incoming
// values for each segment, both directions.
__global__ __launch_bounds__(TPB)
void edt_passB(const float* __restrict__ carryD, const float* __restrict__ carryU,
               float* __restrict__ incD, float* __restrict__ incU) {
    int col = blockIdx.x * TPB + threadIdx.x;
    if (col >= CC) return;

    float c = BIGF;
    for (int s = 0; s < NSEG; ++s) {
        incD[s * CC + col] = c;
        c = fminf(fminf(carryD[s * CC + col], c + (float)SEG), BIGF);
    }
    c = BIGF;
    for (int s = NSEG - 1; s >= 0; --s) {
        incU[s * CC + col] = c;
        c = fminf(fminf(carryU[s * CC + col], c + (float)SEG), BIGF);
    }
}

// ---- Phase 1, pass C: replay segments with correct incoming values,
// combine both directions, emit g^2.
__global__ __launch_bounds__(TPB)
void edt_passC(const float* __restrict__ mask,
               const float* __restrict__ incD, const float* __restrict__ incU,
               float* __restrict__ tmp, float* __restrict__ g2) {
    int col = blockIdx.x * TPB + threadIdx.x;
    int seg = blockIdx.y;
    if (col >= CC) return;
    int s0 = seg * SEG;

    float d = incD[seg * CC + col];
    for (int r = s0; r < s0 + SEG; ++r) {
        float m = mask[r * CC + col];
        d = (m == 0.0f) ? 0.0f : fminf(d + 1.0f, BIGF);
        tmp[r * CC + col] = d;
    }
    float u = incU[seg * CC + col];
    for (int r = s0 + SEG - 1; r >= s0; --r) {
        float m = mask[r * CC + col];
        u = (m == 0.0f) ? 0.0f : fminf(u + 1.0f, BIGF);
        float g = fminf(u, tmp[r * CC + col]);
        g2[r * CC + col] = g * g;
    }
}

// ---- Phase 2: one workgroup per row. Stage g2 row into LDS with gfx1250
// async-to-LDS copies (ASYNCcnt), then exact adaptive outward min-plus search.
__global__ __launch_bounds__(TPB)
void edt_rows(const float* __restrict__ g2, float* __restrict__ out) {
    __shared__ float srow[CC];
    int row = blockIdx.x;
    const float* grow = g2 + (size_t)row * CC;

    // Async copy of the whole 6KB row into LDS (gfx1250 ASYNC path).
    for (int i = threadIdx.x; i < CC; i += TPB) {
        unsigned loff = (unsigned)(uintptr_t)(&srow[i]);  // low 32 bits = LDS addr
        const float* ga = grow + i;
        asm volatile("global_load_async_to_lds_b32 %0, %1, off"
                     :: "v"(loff), "v"(ga) : "memory");
    }
    asm volatile("s_wait_asynccnt 0" ::: "memory");  // my wave's asyncs done
    __syncthreads();                                 // all waves' data visible

    for (int j = threadIdx.x; j < CC; j += TPB) {
        float best = srow[j];            // k == j term
        // Exact: any k with (j-k)^2 >= best cannot improve the minimum.
        float o2 = 1.0f;                 // off^2, maintained incrementally
        float stp = 3.0f;                // (off+1)^2 - off^2 = 2*off + 1
        int off = 1;
        while (o2 < best) {
            float lo = (j - off >= 0) ? srow[j - off] : INF_F;
            float hi = (j + off < CC) ? srow[j + off] : INF_F;
            best = fminf(best, fminf(lo, hi) + o2);
            o2 += stp;
            stp += 2.0f;
            ++off;
        }
        out[(size_t)row * CC + j] = sqrtf(best);
    }
}

extern "C" void kernel_launch(void* const* d_in, const int* in_sizes, int n_in,
                              void* d_out, int out_size, void* d_ws, size_t ws_size,
                              hipStream_t stream) {
    const float* mask = (const float*)d_in[0];
    float* out = (float*)d_out;
    float* ws = (float*)d_ws;

    // Workspace layout (floats): ~20 MB total
    float* tmp    = ws;                        // RR*CC  (down-distance)
    float* g2     = ws + (size_t)RR * CC;      // RR*CC  (combined squared)
    float* carryD = g2 + (size_t)RR * CC;      // NSEG*CC
    float* carryU = carryD + (size_t)NSEG * CC; // NSEG*CC
    float* incD   = carryU + (size_t)NSEG * CC; // NSEG*CC
    float* incU   = incD + (size_t)NSEG * CC;   // NSEG*CC

    dim3 gridSeg(CC / TPB, NSEG);
    edt_passA<<<gridSeg, TPB, 0, stream>>>(mask, carryD, carryU);
    edt_passB<<<CC / TPB, TPB, 0, stream>>>(carryD, carryU, incD, incU);
    edt_passC<<<gridSeg, TPB, 0, stream>>>(mask, incD, incU, tmp, g2);
    edt_rows<<<RR, TPB, 0, stream>>>(g2, out);
}